// GcnNet_58308476011045
// MI455X (gfx1250) — compile-verified
//
#include <hip/hip_runtime.h>
#include <hip/hip_bf16.h>

typedef __attribute__((ext_vector_type(2))) float v2f;
typedef __attribute__((ext_vector_type(8))) float v8f;

#define N_NODES 100000
#define HID     128
#define N_EDGES 1600000
#define BAG     32
#define LABELS  512

// ---------------------------------------------------------------------------
// Zero-fill (agg/res accumulators must start at 0 each call; ws is poisoned)
// ---------------------------------------------------------------------------
__global__ void zero_fill_kernel(float* __restrict__ p, size_t n) {
    size_t i = (size_t)blockIdx.x * blockDim.x + threadIdx.x;
    size_t stride = (size_t)gridDim.x * blockDim.x;
    for (; i < n; i += stride) p[i] = 0.0f;
}

// ---------------------------------------------------------------------------
// EmbeddingBag(sum) + bias + ReLU. One wave32 per node, one float4 per lane.
// ---------------------------------------------------------------------------
__global__ __launch_bounds__(256) void embed_bag_kernel(
        const int* __restrict__ indices, const float* __restrict__ emb,
        const float* __restrict__ bias, float* __restrict__ h) {
    int t    = blockIdx.x * 256 + threadIdx.x;
    int node = t >> 5;
    int lane = t & 31;
    if (node >= N_NODES) return;

    // each lane loads one of the 32 bag indices; broadcast with shuffles
    int myidx = indices[(size_t)node * BAG + lane];

    float4 acc = make_float4(0.f, 0.f, 0.f, 0.f);
#pragma unroll
    for (int j = 0; j < BAG; ++j) {
        int idx  = __shfl(myidx, j, 32);
        float4 v = ((const float4*)(emb + (size_t)idx * HID))[lane];
        acc.x += v.x; acc.y += v.y; acc.z += v.z; acc.w += v.w;
    }
    float4 b = ((const float4*)bias)[lane];
    acc.x = fmaxf(acc.x + b.x, 0.f);
    acc.y = fmaxf(acc.y + b.y, 0.f);
    acc.z = fmaxf(acc.z + b.z, 0.f);
    acc.w = fmaxf(acc.w + b.w, 0.f);
    ((float4*)(h + (size_t)node * HID))[lane] = acc;
}

// ---------------------------------------------------------------------------
// Edge gather + weighted scatter-add into agg (ppi) and res (self).
// One wave32 per edge; float4 per lane; relaxed agent-scope f32 atomics
// (resolve in the 192MB L2 since the 154MB working set is L2-resident).
// ---------------------------------------------------------------------------
__global__ __launch_bounds__(256) void edge_scatter_kernel(
        const float* __restrict__ h,
        const int* __restrict__ src, const int* __restrict__ dst,
        const float* __restrict__ ppi, const float* __restrict__ selfw,
        float* __restrict__ agg, float* __restrict__ res) {
    int t    = blockIdx.x * 256 + threadIdx.x;
    int e    = t >> 5;
    int lane = t & 31;
    if (e >= N_EDGES) return;

    int   s  = src[e];
    int   d  = dst[e];
    float wp = ppi[e];
    float wr = selfw[e];

    float4 v = ((const float4*)(h + (size_t)s * HID))[lane];
    float* ap = agg + (size_t)d * HID + lane * 4;
    float* rp = res + (size_t)d * HID + lane * 4;

    __hip_atomic_fetch_add(ap + 0, v.x * wp, __ATOMIC_RELAXED, __HIP_MEMORY_SCOPE_AGENT);
    __hip_atomic_fetch_add(ap + 1, v.y * wp, __ATOMIC_RELAXED, __HIP_MEMORY_SCOPE_AGENT);
    __hip_atomic_fetch_add(ap + 2, v.z * wp, __ATOMIC_RELAXED, __HIP_MEMORY_SCOPE_AGENT);
    __hip_atomic_fetch_add(ap + 3, v.w * wp, __ATOMIC_RELAXED, __HIP_MEMORY_SCOPE_AGENT);
    __hip_atomic_fetch_add(rp + 0, v.x * wr, __ATOMIC_RELAXED, __HIP_MEMORY_SCOPE_AGENT);
    __hip_atomic_fetch_add(rp + 1, v.y * wr, __ATOMIC_RELAXED, __HIP_MEMORY_SCOPE_AGENT);
    __hip_atomic_fetch_add(rp + 2, v.z * wr, __ATOMIC_RELAXED, __HIP_MEMORY_SCOPE_AGENT);
    __hip_atomic_fetch_add(rp + 3, v.w * wr, __ATOMIC_RELAXED, __HIP_MEMORY_SCOPE_AGENT);
}

// ---------------------------------------------------------------------------
// FP32 WMMA GEMM:  out[M,N] = A[M,128] @ W[128,N] (+bias) [+relu +res]
// One 16x16 tile per wave via V_WMMA_F32_16X16X4_F32, K=128 in 32 steps.
// Block = 256 threads = 8 waves covering 8 consecutive n-tiles of one m-tile.
// A-tile (16x128 = 8KB) staged once in LDS, shared by all 8 waves.
//
// N is a compile-time constant so every unrolled B access is a literal
// 24-bit instruction offset from one per-lane base pointer (no per-iteration
// 64-bit address math), and every A access is a ds_load_b64 literal offset.
//
// Per-lane layouts (wave32, ISA 7.12.2):
//   A 16x4:  VGPR0/1 hold K={0,1} for lanes 0-15, K={2,3} for lanes 16-31; M=lane%16
//   B 4x16:  mirrored (K striped the same way), N = lane%16
//   C/D:     VGPR r: M = r + 8*(lane>=16), N = lane%16
// ---------------------------------------------------------------------------
template <int N, bool FUSE_RELU_RES>
__global__ __launch_bounds__(256) void wmma_gemm_kernel(
        const float* __restrict__ A, const float* __restrict__ W,
        const float* __restrict__ bias, const float* __restrict__ res,
        float* __restrict__ out) {
    __shared__ float As[16 * HID];

    int m0 = blockIdx.x * 16;

    // cooperative A-tile load: 2048 floats = 512 float4, 256 threads x 2
    {
        const float4* Ag  = (const float4*)(A + (size_t)m0 * HID);
        float4*       Asv = (float4*)As;
        Asv[threadIdx.x]       = Ag[threadIdx.x];
        Asv[threadIdx.x + 256] = Ag[threadIdx.x + 256];
    }
    __syncthreads();

    int wave = threadIdx.x >> 5;
    int lane = threadIdx.x & 31;
    int half = lane >> 4;        // 0: lanes 0-15, 1: lanes 16-31
    int mrow = lane & 15;        // M index within tile (for A)
    int n0   = (blockIdx.y * 8 + wave) * 16;
    int nc   = n0 + (lane & 15); // N column this lane owns for B / C / D

    v8f c = {0.f, 0.f, 0.f, 0.f, 0.f, 0.f, 0.f, 0.f};

    // per-lane bases; inner-loop offsets are all compile-time constants
    const float* Bp = W + nc + 2 * half * N;      // B element (K=2*half, N=nc)
    const float* Ap = As + mrow * HID + 2 * half; // A element (M=mrow, K=2*half)

#pragma unroll
    for (int k = 0; k < HID; k += 4) {
        v2f a = *(const v2f*)(Ap + k);            // ds_load_b64, literal offset
        v2f b;
        b.x = Bp[k * N];                          // global_load_b32, literal offset
        b.y = Bp[k * N + N];
        c = __builtin_amdgcn_wmma_f32_16x16x4_f32(
                /*neg_a=*/false, a, /*neg_b=*/false, b,
                /*c_mod=*/(short)0, c, /*reuse_a=*/false, /*reuse_b=*/false);
    }

    float bb = bias[nc];
    // writeback base: element (row = m0 + 8*half, col = nc); rows advance by N
    int base = (m0 + 8 * half) * N + nc;
#pragma unroll
    for (int r = 0; r < 8; ++r) {
        int   off = base + r * N;
        float v   = c[r] + bb;
        if constexpr (FUSE_RELU_RES) {
            v = fmaxf(v, 0.f) + res[off];
        }
        out[off] = v;
    }
}

// ---------------------------------------------------------------------------
// Host-side orchestration
// ---------------------------------------------------------------------------
extern "C" void kernel_launch(void* const* d_in, const int* in_sizes, int n_in,
                              void* d_out, int out_size, void* d_ws, size_t ws_size,
                              hipStream_t stream) {
    (void)in_sizes; (void)n_in; (void)out_size; (void)ws_size;

    const int*   indices = (const int*)  d_in[0];
    // d_in[1] = offsets: bags are fixed-size 32 by construction -> unused
    const int*   src     = (const int*)  d_in[2];
    const int*   dst     = (const int*)  d_in[3];
    const float* ppi     = (const float*)d_in[4];
    const float* selfw   = (const float*)d_in[5];
    const float* emb     = (const float*)d_in[6];
    const float* ibias   = (const float*)d_in[7];
    const float* Ws      = (const float*)d_in[8];   // [2,128,128]
    const float* bs      = (const float*)d_in[9];   // [2,128]
    const float* Wout    = (const float*)d_in[10];  // [128,512]
    const float* bout    = (const float*)d_in[11];  // [512]
    float*       out     = (float*)d_out;

    const size_t SZ = (size_t)N_NODES * HID;   // 12.8M floats = 51.2 MB
    float* bufA = (float*)d_ws;                // h (layer input)
    float* bufB = bufA + SZ;                   // agg / res ping-pong
    float* bufC = bufB + SZ;

    const int edge_blocks  = (N_EDGES * 32) / 256;   // 200000
    const int embed_blocks = (N_NODES * 32) / 256;   // 12500
    const int m_tiles      = N_NODES / 16;           // 6250

    // 1) EmbeddingBag + bias + ReLU -> bufA
    embed_bag_kernel<<<embed_blocks, 256, 0, stream>>>(indices, emb, ibias, bufA);

    // 2) GCN layer 0: h=bufA -> agg=bufB, res=bufC; h_new written over bufC
    zero_fill_kernel<<<8192, 256, 0, stream>>>(bufB, 2 * SZ);
    edge_scatter_kernel<<<edge_blocks, 256, 0, stream>>>(bufA, src, dst, ppi, selfw, bufB, bufC);
    wmma_gemm_kernel<HID, true><<<dim3(m_tiles, 1), 256, 0, stream>>>(
        bufB, Ws, bs, bufC, bufC);

    // 3) GCN layer 1: h=bufC -> agg=bufA, res=bufB; h_new written over bufB
    zero_fill_kernel<<<8192, 256, 0, stream>>>(bufA, 2 * SZ);
    edge_scatter_kernel<<<edge_blocks, 256, 0, stream>>>(bufC, src, dst, ppi, selfw, bufA, bufB);
    wmma_gemm_kernel<HID, true><<<dim3(m_tiles, 1), 256, 0, stream>>>(
        bufA, Ws + HID * HID, bs + HID, bufB, bufB);

    // 4) Output projection: out = bufB @ Wout + bout   (N=512 -> 4 n-tile groups)
    wmma_gemm_kernel<LABELS, false><<<dim3(m_tiles, LABELS / 128), 256, 0, stream>>>(
        bufB, Wout, bout, nullptr, out);
}